// QKV_15942918602939
// MI455X (gfx1250) — compile-verified
//
#include <hip/hip_runtime.h>
#include <cstdint>

#define B_SZ      512
#define D_SZ      256
#define NUM_FIXED 512
#define MAX_VARS  128
#define L_TOT     (NUM_FIXED + 2 * MAX_VARS)   // 768
#define SCALE     0.0625f                      // 1/sqrt(256)

// TDM pads LDS rows: 4 DWORDs of pad every 256 DWORDs -> row stride 260 floats.
// ds_load_b64 fragment reads then hit banks (4*l16 + c) mod 64: conflict-free.
#define LDS_STRIDE 260

typedef __attribute__((ext_vector_type(2))) float    v2f;
typedef __attribute__((ext_vector_type(8))) float    v8f;
typedef __attribute__((ext_vector_type(4))) unsigned v4u;
typedef __attribute__((ext_vector_type(4))) int      v4i;
typedef __attribute__((ext_vector_type(8))) int      v8i;

// ---------------------------------------------------------------------------
// Issue a 2-D TENSOR_LOAD_TO_LDS: tile_rows x 256 f32, row stride 256 f32 in
// memory, padded rows (stride 260 f32) in LDS. D# bit layout per ISA §8.3-8.5.
// ---------------------------------------------------------------------------
__device__ __forceinline__ void tdm_load_2d(unsigned lds_off, const float* gptr,
                                            unsigned tile_rows) {
  unsigned long long ga = (unsigned long long)(uintptr_t)gptr;
  v4u g0;
  g0.x = 1u;                                   // count=1 (valid user descriptor)
  g0.y = lds_off;                              // lds_addr (bytes)
  g0.z = (unsigned)ga;                         // global_addr[31:0]
  g0.w = (unsigned)((ga >> 32) & 0x1FFFFFFu)   // global_addr[56:32]
       | (2u << 30);                           // type = 2 ("image")
  v8i g1;
  g1[0] = (int)((2u << 16)        // data_size = 4 bytes
              | (1u << 20)        // pad_enable
              | (7u << 22)        // pad_interval: 256 DWORDs
              | (3u << 25));      // pad_amount:   4 DWORDs
  g1[1] = (int)(256u << 16);      // tensor_dim0 = 256 (bits 79:48, low half)
  g1[2] = (int)(tile_rows << 16); // tensor_dim0 hi=0 | tensor_dim1 = tile_rows
  g1[3] = (int)(256u << 16);      // tensor_dim1 hi=0 | tile_dim0 = 256
  g1[4] = (int)tile_rows;         // tile_dim1 | tile_dim2 = 0
  g1[5] = 256;                    // tensor_dim0_stride = 256 (low 32 of 48)
  g1[6] = 0;
  g1[7] = 0;
  v4i gz = {0, 0, 0, 0};
#if defined(__clang_major__) && (__clang_major__ >= 23)
  v8i gz8 = {0, 0, 0, 0, 0, 0, 0, 0};
  __builtin_amdgcn_tensor_load_to_lds(g0, g1, gz, gz, gz8, 0);
#else
  __builtin_amdgcn_tensor_load_to_lds(g0, g1, gz, gz, 0);
#endif
}

// ---------------------------------------------------------------------------
// Kernel 1: fixed-key GEMM out[0:512, 0:512] = q * k^T via WMMA f32 16x16x4.
// Each block: one 16-row q tile x one 128-row k tile, both DMA'd to LDS by
// the Tensor Data Mover (wave 0), consumed by 8 waves (one 16x16 tile each)
// through conflict-free ds_load_b64 fragment reads.
// ---------------------------------------------------------------------------
__global__ __launch_bounds__(256) void attn_fixed_wmma(
    const float* __restrict__ q, const float* __restrict__ kfix,
    float* __restrict__ out)
{
  __shared__ __align__(16) float lds_q[16  * LDS_STRIDE];   // ~16.6 KB
  __shared__ __align__(16) float lds_k[128 * LDS_STRIDE];   // ~133 KB (<=320KB/WGP)

  const int lane = threadIdx.x & 31;
  const int wave = threadIdx.x >> 5;
  const int bm   = blockIdx.x >> 2;   // q-row tile   (0..31)
  const int bn   = blockIdx.x & 3;    // 128-col group (0..3)

  if (wave == 0) {
    tdm_load_2d((unsigned)(uintptr_t)&lds_q[0], q    + (size_t)bm * 16  * D_SZ, 16);
    tdm_load_2d((unsigned)(uintptr_t)&lds_k[0], kfix + (size_t)bn * 128 * D_SZ, 128);
    __builtin_amdgcn_s_wait_tensorcnt(0);
  }
  __syncthreads();

  // Fragment addressing (ISA 7.12.2, 32-bit A 16x4 / B 4x16):
  // lane<16: row=lane, K=k0..k0+1 ; lane>=16: row=lane-16, K=k0+2..k0+3.
  const int half = lane >> 4;
  const int l16  = lane & 15;
  const float* qrow = &lds_q[l16 * LDS_STRIDE + 2 * half];
  const float* krow = &lds_k[(wave * 16 + l16) * LDS_STRIDE + 2 * half];

  v8f acc = {};
#pragma unroll 8
  for (int kk = 0; kk < D_SZ; kk += 4) {
    v2f a = *(const v2f*)(qrow + kk);
    v2f b = *(const v2f*)(krow + kk);
    // 8 args: (neg_a, A, neg_b, B, c_mod, C, reuse_a, reuse_b)
    acc = __builtin_amdgcn_wmma_f32_16x16x4_f32(
        false, a, false, b, (short)0, acc, false, false);
  }

  // C/D layout: VGPR v -> (M = v + 8*half, N = l16) within the 16x16 tile.
  float* orow = out + (size_t)(bm * 16) * L_TOT + bn * 128 + wave * 16 + l16;
#pragma unroll
  for (int v = 0; v < 8; ++v)
    orow[(size_t)(v + 8 * half) * L_TOT] = acc[v] * SCALE;
}

// ---------------------------------------------------------------------------
// Kernel 2: k_var + gathered k_arg columns (the ~128 MB HBM stream that
// dominates runtime; no cross-batch reuse -> WMMA cannot help). One block per
// batch element; each wave processes 4 key rows per iteration so 8 coalesced
// b128 loads are in flight before the dependent reductions, and the 4
// contiguous results are written with one global_store_b128.
// ---------------------------------------------------------------------------
__global__ __launch_bounds__(256) void attn_var(
    const float* __restrict__ q, const float* __restrict__ kvar,
    const int* __restrict__ args, const float* __restrict__ karg,
    float* __restrict__ out)
{
  const int b    = blockIdx.x;
  const int lane = threadIdx.x & 31;
  const int wave = threadIdx.x >> 5;

  // Each lane keeps its private 8-float slice of q[b] in registers.
  const float4 q0 = *(const float4*)(q + (size_t)b * D_SZ + lane * 8);
  const float4 q1 = *(const float4*)(q + (size_t)b * D_SZ + lane * 8 + 4);

  for (int m0 = wave * 4; m0 < 2 * MAX_VARS; m0 += 32) {
    // Resolve 4 row pointers (wave-uniform; sentinel gather rows -> nullptr).
    const float* rows[4];
#pragma unroll
    for (int r = 0; r < 4; ++r) {
      const int m = m0 + r;
      if (m < MAX_VARS) {
        rows[r] = kvar + ((size_t)b * MAX_VARS + m) * D_SZ;
      } else {
        const int mm = m - MAX_VARS;
        const int i = args[((size_t)b * MAX_VARS + mm) * 2 + 0];
        const int j = args[((size_t)b * MAX_VARS + mm) * 2 + 1];
        rows[r] = (i < 0) ? nullptr
                          : karg + (size_t)(i * MAX_VARS + j) * D_SZ;
      }
    }

    // Prefetch the k_var stream one wave-iteration ahead (global_prefetch_b8).
    if (m0 + 32 < MAX_VARS)
      __builtin_prefetch(rows[0] + 32 * D_SZ, 0, 3);

    // 8 independent b128 loads + dot partials.
    float s[4];
#pragma unroll
    for (int r = 0; r < 4; ++r) {
      if (rows[r]) {
        const float4 k0 = *(const float4*)(rows[r] + lane * 8);
        const float4 k1 = *(const float4*)(rows[r] + lane * 8 + 4);
        s[r] = q0.x * k0.x + q0.y * k0.y + q0.z * k0.z + q0.w * k0.w
             + q1.x * k1.x + q1.y * k1.y + q1.z * k1.z + q1.w * k1.w;
      } else {
        s[r] = 0.0f;   // masked sentinel row -> exact zero
      }
    }

    // 4 interleaved wave32 butterfly reductions.
#pragma unroll
    for (int off = 16; off > 0; off >>= 1) {
#pragma unroll
      for (int r = 0; r < 4; ++r)
        s[r] += __shfl_xor(s[r], off, 32);
    }

    if (lane == 0) {
      float4 o = make_float4(s[0] * SCALE, s[1] * SCALE,
                             s[2] * SCALE, s[3] * SCALE);
      *(float4*)(out + (size_t)b * L_TOT + NUM_FIXED + m0) = o;  // b128 store
    }
  }
}

extern "C" void kernel_launch(void* const* d_in, const int* in_sizes, int n_in,
                              void* d_out, int out_size, void* d_ws, size_t ws_size,
                              hipStream_t stream) {
  const float* q    = (const float*)d_in[0];   // [512, 256]
  const float* kvar = (const float*)d_in[1];   // [512, 128, 256]
  const int*   args = (const int*)d_in[2];     // [512, 128, 2]
  const float* kfix = (const float*)d_in[3];   // [512, 256]
  const float* karg = (const float*)d_in[4];   // [16384, 256]
  float* out = (float*)d_out;                  // [512, 768]

  // 32 q-row tiles x 4 column groups of 128 -> 128 blocks, 8 waves each.
  attn_fixed_wmma<<<128, 256, 0, stream>>>(q, kfix, out);
  // One block per batch element for the streaming/gather columns.
  attn_var<<<B_SZ, 256, 0, stream>>>(q, kvar, args, karg, out);
}